// OctreeDWConv_6777458393267
// MI455X (gfx1250) — compile-verified
//
#include <hip/hip_runtime.h>

// out[i,c] = sum_k (neigh[i,k] >= 0 ? data[neigh[i,k], c] : 0) * w[k, c]
// N = 200000, K = 27, C = 64, fp32. Gather-bound; data (51 MB) is L2-resident
// (192 MB L2). HBM floor ~124 MB => ~5.3 us. Compute (0.69 GFLOP) negligible.

typedef float float4v __attribute__((ext_vector_type(4)));
typedef int   v4i_t   __attribute__((__vector_size__(16)));   // matches builtin proto

#define K_NBR 27
#define C_CH 64
#define C4 (C_CH / 4)          // 16 channel-quads
#define PTS_PER_BLOCK 16
#define THREADS 256

// ---- CDNA5 async Global->LDS staging (ASYNCcnt path), guarded by __has_builtin ----
#if __has_builtin(__builtin_amdgcn_global_load_async_to_lds_b128) && \
    __has_builtin(__builtin_amdgcn_global_load_async_to_lds_b32)  && \
    __has_builtin(__builtin_amdgcn_s_wait_asynccnt)
#define USE_ASYNC_LDS 1
#else
#define USE_ASYNC_LDS 0
#endif

#if USE_ASYNC_LDS
typedef __attribute__((address_space(1))) int   as1_int;
typedef __attribute__((address_space(3))) int   as3_int;
typedef __attribute__((address_space(1))) v4i_t as1_v4i;
typedef __attribute__((address_space(3))) v4i_t as3_v4i;

// Generic->AS1: identical bit pattern (flat address == global address).
// Generic->AS3: low 32 bits of a flat LDS pointer are the LDS byte offset
// (flat aperture mapping, ISA 10.2).
__device__ __forceinline__ void async_lds_b32(const int* g, const int* l) {
    __builtin_amdgcn_global_load_async_to_lds_b32(
        (as1_int*)(uintptr_t)g, (as3_int*)(unsigned)(uintptr_t)l, 0, 0);
}
__device__ __forceinline__ void async_lds_b128(const void* g, const void* l) {
    __builtin_amdgcn_global_load_async_to_lds_b128(
        (as1_v4i*)(uintptr_t)g, (as3_v4i*)(unsigned)(uintptr_t)l, 0, 0);
}
#endif

__global__ __launch_bounds__(THREADS)
void octree_dwconv_kernel(const float* __restrict__ data,
                          const float* __restrict__ weights,
                          const int* __restrict__ neigh,
                          float* __restrict__ out,
                          int n_points) {
    __shared__ int   s_neigh[PTS_PER_BLOCK * K_NBR];  // 432 ints = 1728 B
    __shared__ float s_w[K_NBR * C_CH];               // 6912 B = 432 x b128

    const int t  = threadIdx.x;
    const int c4 = t & (C4 - 1);   // channel quad 0..15
    const int p  = t >> 4;         // point slot   0..15
    const int i0 = blockIdx.x * PTS_PER_BLOCK;

    const int pts  = (n_points - i0) < PTS_PER_BLOCK ? (n_points - i0) : PTS_PER_BLOCK;
    const int tile = pts * K_NBR;

    const int* nt = neigh + (long)i0 * K_NBR;

#if USE_ASYNC_LDS
    // --- async DMA staging: Global -> LDS, tracked by ASYNCcnt (no VGPR round trip) ---
    // neigh tile: b32 granularity (exact for partial tail tiles)
    for (int idx = t; idx < tile; idx += THREADS)
        async_lds_b32(nt + idx, &s_neigh[idx]);
    // weight matrix: 432 b128 chunks
    for (int idx = t; idx < (K_NBR * C_CH) / 4; idx += THREADS)
        async_lds_b128(weights + idx * 4, &s_w[idx * 4]);
    __builtin_amdgcn_s_wait_asynccnt(0);   // drain this wave's async copies
#else
    // --- fallback: cooperative staging through VGPRs ---
    for (int idx = t; idx < tile; idx += THREADS)
        s_neigh[idx] = __builtin_nontemporal_load(nt + idx);
    for (int idx = t; idx < K_NBR * C_CH; idx += THREADS)
        s_w[idx] = weights[idx];
#endif
    __syncthreads();

    if (p < pts) {
        const int i = i0 + p;
        const float4v* __restrict__ data4 = reinterpret_cast<const float4v*>(data);
        const float4v* __restrict__ sw4   = reinterpret_cast<const float4v*>(s_w);

        // Phase 1: compute all 27 gather addresses + validity, issue all loads
        // back-to-back (b128 each; 16 threads/row => one coalesced 256B row).
        float4v d[K_NBR];
        float   m[K_NBR];
#pragma unroll
        for (int k = 0; k < K_NBR; ++k) {
            const int n = s_neigh[p * K_NBR + k];     // LDS broadcast within group
            m[k] = (n >= 0) ? 1.0f : 0.0f;            // branchless invalid mask
            const int nc = (n >= 0) ? n : 0;          // clamp (data finite, masked out)
            d[k] = data4[nc * C4 + c4];               // global_load_b128 gather
        }

        // Phase 2: depthwise reduction. Weight quads from LDS: group of 16
        // threads reads 16 distinct b128 words -> all 64 banks, conflict-free.
        float4v acc = {0.f, 0.f, 0.f, 0.f};
#pragma unroll
        for (int k = 0; k < K_NBR; ++k) {
            const float4v wk = sw4[k * C4 + c4];
            acc += d[k] * (wk * m[k]);
        }

        // Streamed output, never re-read -> nontemporal store (keep L2 for data).
        float4v* out4 = reinterpret_cast<float4v*>(out);
        __builtin_nontemporal_store(acc, &out4[i * C4 + c4]);
    }
}

extern "C" void kernel_launch(void* const* d_in, const int* in_sizes, int n_in,
                              void* d_out, int out_size, void* d_ws, size_t ws_size,
                              hipStream_t stream) {
    const float* data    = (const float*)d_in[0];   // [N, 64]
    const float* weights = (const float*)d_in[1];   // [27, 1, 64] flat
    const int*   neigh   = (const int*)d_in[2];     // [N, 27]
    float*       out     = (float*)d_out;           // [N, 64]

    const int n_points = in_sizes[0] / C_CH;        // 200000
    const int grid = (n_points + PTS_PER_BLOCK - 1) / PTS_PER_BLOCK;

    octree_dwconv_kernel<<<grid, THREADS, 0, stream>>>(data, weights, neigh, out, n_points);
}